// TripletLoss2_10393820856648
// MI455X (gfx1250) — compile-verified
//
#include <hip/hip_runtime.h>

// ---------------------------------------------------------------------------
// TripletLoss2 for MI455X (gfx1250, wave32).
//
// Phase 1 (bandwidth-bound, the real work): one wave per row.
//   reads 3 rows * 4096 f32 via global_load_b128 (float4), accumulates
//   s1 = sum((f[r]-f[idx1])^2), s2 = sum((f[r]-f[idx2])^2) per lane,
//   shuffle-reduces across the 32 lanes, lane 0 applies the hinge formula
//   and writes one float per row into d_ws.
//   192 MB touched; f fits in 192MB L2 so gathers mostly hit L2.
//
// Phase 2 (deterministic scalar reduction on the matrix pipe): one wave.
//   V_WMMA_F32_16X16X4_F32 with A = ones(16x4): D[m][n] += sum_k B[k][n].
//   Each iteration feeds 64 floats (a full wave32 B operand = 2 VGPRs) and
//   accumulates column sums in C. Summing the 16 columns of row 0 at the end
//   yields the exact f32 total with a fixed accumulation order.
// ---------------------------------------------------------------------------

typedef __attribute__((ext_vector_type(2))) float v2f;
typedef __attribute__((ext_vector_type(8))) float v8f;

#define MARGIN 0.5f

__global__ __launch_bounds__(256) void triplet_rows_kernel(
    const float* __restrict__ f,
    const float* __restrict__ label,
    const int*   __restrict__ idx1,
    const int*   __restrict__ idx2,
    float*       __restrict__ row_hinge,
    int D)
{
    const int wave = threadIdx.x >> 5;   // 8 waves per block
    const int lane = threadIdx.x & 31;
    const int r    = blockIdx.x * 8 + wave;

    const int i1 = idx1[r];
    const int i2 = idx2[r];

    const float4* __restrict__ fa = (const float4*)(f + (size_t)r  * D);
    const float4* __restrict__ fb = (const float4*)(f + (size_t)i1 * D);
    const float4* __restrict__ fc = (const float4*)(f + (size_t)i2 * D);

    float s1 = 0.0f, s2 = 0.0f;

    // D/4 float4 per row; 32 lanes -> D/128 iterations (D=4096 -> 32 iters).
    const int nIt = D >> 7;
#pragma unroll 4
    for (int it = 0; it < nIt; ++it) {
        const int p = it * 32 + lane;     // consecutive lanes -> consecutive 16B
        float4 a = fa[p];
        float4 b = fb[p];
        float4 c = fc[p];

        float d0 = a.x - b.x, d1 = a.y - b.y, d2 = a.z - b.z, d3 = a.w - b.w;
        s1 = fmaf(d0, d0, s1); s1 = fmaf(d1, d1, s1);
        s1 = fmaf(d2, d2, s1); s1 = fmaf(d3, d3, s1);

        float e0 = a.x - c.x, e1 = a.y - c.y, e2 = a.z - c.z, e3 = a.w - c.w;
        s2 = fmaf(e0, e0, s2); s2 = fmaf(e1, e1, s2);
        s2 = fmaf(e2, e2, s2); s2 = fmaf(e3, e3, s2);
    }

    // wave32 butterfly reduction
#pragma unroll
    for (int off = 16; off > 0; off >>= 1) {
        s1 += __shfl_xor(s1, off, 32);
        s2 += __shfl_xor(s2, off, 32);
    }

    if (lane == 0) {
        const float lr = label[r];
        const float l1 = label[i1];
        const float l2 = label[i2];
        const float d1 = fabsf(lr - l1);
        const float d2 = fabsf(lr - l2);
        const bool  cond  = (d1 >= d2);     // true -> idx2 is the 'near' sample
        const float a2n   = cond ? s2 : s1;
        const float a2f   = cond ? s1 : s2;
        const float nearl = cond ? l2 : l1;
        const float farl  = cond ? l1 : l2;
        const float alpha = (lr - farl) * (lr - farl) - (lr - nearl) * (lr - nearl);
        const float loss  = a2n - a2f + alpha * MARGIN;
        row_hinge[r] = fmaxf(loss, 0.0f);
    }
}

// Single-wave WMMA reduction of N floats (N % 64 == 0) to a scalar.
// A = ones(16x4) => D[m][n] = sum_k B[k][n] accumulated into C across tiles.
// Every (lane, vgpr) slot of B maps to a unique (k,n), so summing all 16
// column sums at the end equals the sum of every element fed in.
__global__ __launch_bounds__(32) void reduce_wmma_kernel(
    const float* __restrict__ in,
    float*       __restrict__ out,
    int N)
{
    const int lane = threadIdx.x;   // 0..31, one full wave (EXEC all ones)

    v8f acc = {};
    v2f ones;
    ones[0] = 1.0f;
    ones[1] = 1.0f;

    const int nIt = N >> 6;         // 64 values per WMMA (4x16 B tile)
    for (int it = 0; it < nIt; ++it) {
        v2f b;
        b[0] = in[it * 64 + lane];
        b[1] = in[it * 64 + 32 + lane];
        acc = __builtin_amdgcn_wmma_f32_16x16x4_f32(
            /*neg_a=*/false, ones,
            /*neg_b=*/false, b,
            /*c_mod=*/(short)0, acc,
            /*reuse_a=*/false, /*reuse_b=*/false);
    }

    // acc[0] on lanes 0..15 holds D[0][n] = column sums n = lane.
    float col = acc[0];
    col += __shfl_xor(col, 8, 32);
    col += __shfl_xor(col, 4, 32);
    col += __shfl_xor(col, 2, 32);
    col += __shfl_xor(col, 1, 32);

    if (lane == 0) out[0] = col;
}

extern "C" void kernel_launch(void* const* d_in, const int* in_sizes, int n_in,
                              void* d_out, int out_size, void* d_ws, size_t ws_size,
                              hipStream_t stream) {
    const float* feats = (const float*)d_in[0];
    const float* label = (const float*)d_in[1];
    const int*   idx1  = (const int*)d_in[2];
    const int*   idx2  = (const int*)d_in[3];
    float* out = (float*)d_out;

    const int B = in_sizes[1];              // 4096
    const int D = in_sizes[0] / B;          // 4096

    float* row_hinge = (float*)d_ws;        // B floats of scratch

    // Phase 1: 8 rows (waves) per 256-thread block.
    triplet_rows_kernel<<<B / 8, 256, 0, stream>>>(feats, label, idx1, idx2,
                                                   row_hinge, D);
    // Phase 2: single-wave deterministic WMMA reduction.
    reduce_wmma_kernel<<<1, 32, 0, stream>>>(row_hinge, out, B);
}